// SpanProposal_87849261072686
// MI455X (gfx1250) — compile-verified
//
#include <hip/hip_runtime.h>
#include <math.h>

// ---------------- problem constants (from reference) ----------------
#define B_SZ     8
#define S_LEN    256
#define HD       768
#define RR       256          // R
#define WD       32           // Wd
#define MAX_SPAN 8
#define N_SPANS  2020         // 249*8 + (7+6+5+4+3+2+1)
#define RAW_DIM  2336         // 3*HD + WD
#define NEG_INF  (-1000000000.0f)

// output layout (flat concat in return order)
#define OFF_SR   0
#define OFF_ASP  (B_SZ * N_SPANS * RR)               // 4,136,960
#define OFF_OPN  (OFF_ASP + B_SZ * N_SPANS)          // +16,160
#define OFF_NASP (OFF_OPN + B_SZ * N_SPANS)          // +16,160
#define OFF_NOPN (OFF_NASP + B_SZ * RR)              // +2,048

typedef float v2f __attribute__((ext_vector_type(2)));
typedef float v8f __attribute__((ext_vector_type(8)));

__device__ __forceinline__ float wave_reduce_sum(float v) {
  // wave32 butterfly reduce
  #pragma unroll
  for (int m = 16; m >= 1; m >>= 1) v += __shfl_xor(v, m, 32);
  return v;
}

// async global->LDS copy of 16 bytes per lane (CDNA5 GLOBAL_LOAD_ASYNC_TO_LDS_B128)
// ldsoff: LDS byte offset (relative to workgroup LDS base); gp: global address.
__device__ __forceinline__ void async_copy_b128(unsigned ldsoff, const void* gp) {
  asm volatile("global_load_async_to_lds_b128 %0, %1, off"
               :: "v"(ldsoff), "v"(gp)
               : "memory");
}
__device__ __forceinline__ void wait_asynccnt0() {
  asm volatile("s_wait_asynccnt 0" ::: "memory");
}

// ---------------- kernel 1: logits[b,s] = H[b,s,:]·w_attn + b_attn ----------------
// one wave per row; grid = B*S/8 blocks of 256 threads (8 waves)
__global__ __launch_bounds__(256) void logits_kernel(
    const float* __restrict__ H, const float* __restrict__ w_attn,
    const float* __restrict__ b_attn, float* __restrict__ logits) {
  const int wave = threadIdx.x >> 5;
  const int lane = threadIdx.x & 31;
  const int row  = blockIdx.x * 8 + wave;   // 0 .. B*S-1
  const float* h = H + (size_t)row * HD;
  float acc = 0.f;
  for (int k = lane; k < HD; k += 32) acc += h[k] * w_attn[k];
  acc = wave_reduce_sum(acc);
  if (lane == 0) logits[row] = acc + b_attn[0];
}

// ---------------- kernel 2: masked mean over sequence ----------------
__global__ __launch_bounds__(256) void hmean_kernel(
    const float* __restrict__ H, const float* __restrict__ mask,
    float* __restrict__ hmean) {
  const int b = blockIdx.x;
  const int c = threadIdx.x;           // owns cols c, c+256, c+512
  float a0 = 0.f, a1 = 0.f, a2 = 0.f, len = 0.f;
  for (int s = 0; s < S_LEN; ++s) {
    const float mk = mask[b * S_LEN + s];
    const float* h = H + ((size_t)(b * S_LEN + s)) * HD;
    a0 += mk * h[c];
    a1 += mk * h[c + 256];
    a2 += mk * h[c + 512];
    len += mk;
  }
  len = fmaxf(len, 1.0f);
  hmean[b * HD + c]       = a0 / len;
  hmean[b * HD + c + 256] = a1 / len;
  hmean[b * HD + c + 512] = a2 / len;
}

// ---------------- kernel 3: null heads: tanh(h_mean @ W + b) ----------------
// grid = B*2 blocks; thread r reads W column r (coalesced: W is [HD][RR] row-major)
__global__ __launch_bounds__(256) void null_kernel(
    const float* __restrict__ hmean,
    const float* __restrict__ W_nasp, const float* __restrict__ b_nasp,
    const float* __restrict__ W_nopn, const float* __restrict__ b_nopn,
    float* __restrict__ out_nasp, float* __restrict__ out_nopn) {
  const int b    = blockIdx.x >> 1;
  const int head = blockIdx.x & 1;
  const int r    = threadIdx.x;
  const float* W    = head ? W_nopn : W_nasp;
  const float* bias = head ? b_nopn : b_nasp;
  float*       out  = head ? out_nopn : out_nasp;
  const float* hm = hmean + b * HD;
  float acc = 0.f;
  for (int k = 0; k < HD; ++k) acc += hm[k] * W[k * RR + r];
  out[b * RR + r] = tanhf(acc + bias[r]);
}

// ---------------- kernel 4: main span kernel ----------------
// One workgroup = (batch b) x (tile of 16 spans).  8 wave32 per block.
// A-tile (16 x Kseg) built in LDS per segment (async global->LDS for the copy
// segments); each wave owns 32 output cols (two 16x16 f32 accumulators) and
// runs v_wmma_f32_16x16x4_f32 over K.
#define STRIDE 776    // LDS row stride (floats): >=768, byte stride 3104 = 16B aligned
#define OSTR   264    // relu-staging row stride

__global__ __launch_bounds__(256) void span_kernel(
    const float* __restrict__ H, const float* __restrict__ logits,
    const float* __restrict__ W_width, const float* __restrict__ W_span,
    const float* __restrict__ b_span,
    const float* __restrict__ w_asp, const float* __restrict__ b_asp,
    const float* __restrict__ w_opn, const float* __restrict__ b_opn,
    float* __restrict__ out_sr, float* __restrict__ out_asp,
    float* __restrict__ out_opn) {
  __shared__ __attribute__((aligned(16))) float tile[16 * STRIDE]; // ~48.5 KB
  __shared__ float aw[16][8];
  __shared__ int   s0s[16], wids[16];

  const int b    = blockIdx.y;
  const int n0   = blockIdx.x * 16;
  const int tid  = threadIdx.x;
  const int lane = tid & 31;
  const int wv   = tid >> 5;

  // ---- span table (analytic enumeration) ----
  if (tid < 16) {
    int n  = n0 + tid;
    int nc = (n < N_SPANS) ? n : (N_SPANS - 1);
    int start, width;
    if (nc < 1992) {                       // full blocks of 8 spans
      start = nc >> 3;
      width = (nc & 7) + 1;
    } else {                               // tail: 7,6,...,1 spans
      int r = nc - 1992, i = 0, cnt = 7;
      while (r >= cnt) { r -= cnt; --cnt; ++i; }
      start = 249 + i;
      width = r + 1;
    }
    s0s[tid]  = start;
    wids[tid] = width;
  }
  __syncthreads();

  // ---- per-span softmax over <=8 window logits ----
  if (tid < 16) {
    const int start = s0s[tid], width = wids[tid];
    float v[8], mx = NEG_INF;
    #pragma unroll
    for (int l = 0; l < 8; ++l) {
      float x = (l < width) ? logits[b * S_LEN + start + l] : NEG_INF;
      v[l] = x;
      mx = fmaxf(mx, x);
    }
    float s = 0.f;
    #pragma unroll
    for (int l = 0; l < 8; ++l) { v[l] = __expf(v[l] - mx); s += v[l]; }
    const float inv = 1.0f / s;
    #pragma unroll
    for (int l = 0; l < 8; ++l) aw[tid][l] = v[l] * inv;
  }

  // ---- persistent fp32 WMMA accumulators: wave wv owns cols [32*wv, 32*wv+32) ----
  v8f acc0 = {0.f, 0.f, 0.f, 0.f, 0.f, 0.f, 0.f, 0.f};
  v8f acc1 = {0.f, 0.f, 0.f, 0.f, 0.f, 0.f, 0.f, 0.f};
  const int col0 = wv * 32;
  const int colA = col0 + (lane & 15);
  const int colB = colA + 16;
  const int arow = lane & 15;              // A-frag row for this lane
  const int koff = (lane < 16) ? 0 : 2;    // A/B frag K sub-offset per lane half

  // ---- 4 K-segments: h_start(768) | h_end(768) | pooled(768) | e_width(32) ----
  #pragma unroll 1
  for (int seg = 0; seg < 4; ++seg) {
    __syncthreads();   // previous segment's LDS reads done before rebuild

    if (seg == 0 || seg == 1) {
      // gather-copy H[b, s0|e0, :] rows straight into LDS via the CDNA5
      // async copy engine (16 B per lane per issue, no VGPR round-trip)
      for (int idx = tid; idx < 16 * (HD / 4); idx += 256) {
        const int m  = idx / (HD / 4);
        const int c4 = idx % (HD / 4);
        const int srow = (seg == 0) ? s0s[m] : (s0s[m] + wids[m] - 1);
        const float* gp =
            H + ((size_t)(b * S_LEN + srow)) * HD + c4 * 4;
        const unsigned ldsoff =
            (unsigned)(size_t)(&tile[m * STRIDE + c4 * 4]);
        async_copy_b128(ldsoff, gp);
      }
      wait_asynccnt0();   // ASYNCcnt==0 -> LDS writes landed; barrier publishes
    } else if (seg == 2) {
      // pooled = softmax-weighted sum over window tokens
      for (int idx = tid; idx < 16 * (HD / 4); idx += 256) {
        const int m  = idx / (HD / 4);
        const int c4 = idx % (HD / 4);
        const int start = s0s[m];
        float4 a4 = {0.f, 0.f, 0.f, 0.f};
        #pragma unroll
        for (int l = 0; l < 8; ++l) {
          int srow = start + l;
          if (srow > S_LEN - 1) srow = S_LEN - 1;
          const float w = aw[m][l];
          const float4 hv =
              *(const float4*)(H + ((size_t)(b * S_LEN + srow)) * HD + c4 * 4);
          a4.x += w * hv.x; a4.y += w * hv.y; a4.z += w * hv.z; a4.w += w * hv.w;
        }
        *(float4*)(&tile[m * STRIDE + c4 * 4]) = a4;
      }
    } else {
      // width embedding (16 x 32)
      for (int idx = tid; idx < 16 * WD; idx += 256) {
        const int m = idx >> 5;
        const int c = idx & 31;
        tile[m * STRIDE + c] = W_width[wids[m] * WD + c];
      }
    }
    __syncthreads();

    const int segK    = (seg == 3) ? WD : HD;
    const int segBase = seg * HD;          // seg==3 -> 2304 (3*HD) since WD chunk is last

    // K loop: unroll x2 -> 4 v_wmma_f32_16x16x4_f32 per iteration, letting the
    // scheduler overlap the next load clause with the current WMMAs.
    #pragma unroll 2
    for (int k0 = 0; k0 < segK; k0 += 4) {
      const int kg = segBase + k0 + koff;

      // A fragment: lane<16 -> (m, k0..k0+1), lane>=16 -> (m, k0+2..k0+3); ds_load_b64
      const v2f a = *(const v2f*)&tile[arow * STRIDE + k0 + koff];

      // B fragments from W_span (row-major [RAW_DIM][RR]; K-rows strided by RR)
      const float* wp = W_span + (size_t)kg * RR;
      v2f bA, bB;
      bA.x = wp[colA];       bA.y = wp[RR + colA];
      bB.x = wp[colB];       bB.y = wp[RR + colB];

      // prefetch the B panel 8 K-rows ahead into near caches
      __builtin_prefetch(wp + 8 * RR + colA, 0, 3);

      acc0 = __builtin_amdgcn_wmma_f32_16x16x4_f32(
          false, a, false, bA, (short)0, acc0, false, false);
      acc1 = __builtin_amdgcn_wmma_f32_16x16x4_f32(
          false, a, false, bB, (short)0, acc1, false, false);
    }
  }

  __syncthreads();   // done with A tile; reuse LDS as relu staging

  // ---- epilogue: bias + relu, store span_reprs, stage for score reductions ----
  const float biasA = b_span[colA];
  const float biasB = b_span[colB];
  const int   mAdd  = (lane < 16) ? 0 : 8;   // D layout: lanes 16-31 hold M = v+8
  #pragma unroll
  for (int v = 0; v < 8; ++v) {
    const int m = v + mAdd;
    const float r0 = fmaxf(acc0[v] + biasA, 0.f);
    const float r1 = fmaxf(acc1[v] + biasB, 0.f);
    tile[m * OSTR + colA] = r0;
    tile[m * OSTR + colB] = r1;
    const int n = n0 + m;
    if (n < N_SPANS) {
      const size_t o = ((size_t)(b * N_SPANS + n)) * RR;
      out_sr[o + colA] = r0;
      out_sr[o + colB] = r1;
    }
  }
  __syncthreads();

  // ---- asp/opn scores: wave wv reduces rows wv and wv+8 ----
  #pragma unroll
  for (int mi = 0; mi < 2; ++mi) {
    const int m = wv + mi * 8;
    const int n = n0 + m;
    float sa = 0.f, so = 0.f;
    for (int c = lane; c < RR; c += 32) {
      const float x = tile[m * OSTR + c];
      sa += x * w_asp[c];
      so += x * w_opn[c];
    }
    sa = wave_reduce_sum(sa);
    so = wave_reduce_sum(so);
    if (lane == 0 && n < N_SPANS) {
      out_asp[b * N_SPANS + n] = sa + b_asp[0];
      out_opn[b * N_SPANS + n] = so + b_opn[0];
    }
  }
}

// ---------------- launch ----------------
extern "C" void kernel_launch(void* const* d_in, const int* in_sizes, int n_in,
                              void* d_out, int out_size, void* d_ws, size_t ws_size,
                              hipStream_t stream) {
  const float* H        = (const float*)d_in[0];
  const float* mask     = (const float*)d_in[1];
  const float* W_width  = (const float*)d_in[2];
  const float* w_attn   = (const float*)d_in[3];
  const float* b_attn   = (const float*)d_in[4];
  const float* W_span   = (const float*)d_in[5];
  const float* b_span   = (const float*)d_in[6];
  const float* w_asp    = (const float*)d_in[7];
  const float* b_asp    = (const float*)d_in[8];
  const float* w_opn    = (const float*)d_in[9];
  const float* b_opn    = (const float*)d_in[10];
  const float* W_nasp   = (const float*)d_in[11];
  const float* b_nasp   = (const float*)d_in[12];
  const float* W_nopn   = (const float*)d_in[13];
  const float* b_nopn   = (const float*)d_in[14];

  float* out    = (float*)d_out;
  float* logits = (float*)d_ws;                 // B*S floats
  float* hmean  = logits + B_SZ * S_LEN;        // B*HD floats

  logits_kernel<<<(B_SZ * S_LEN) / 8, 256, 0, stream>>>(H, w_attn, b_attn, logits);
  hmean_kernel<<<B_SZ, 256, 0, stream>>>(H, mask, hmean);
  null_kernel<<<B_SZ * 2, 256, 0, stream>>>(hmean, W_nasp, b_nasp, W_nopn, b_nopn,
                                            out + OFF_NASP, out + OFF_NOPN);

  const int ntiles = (N_SPANS + 15) / 16;       // 127
  span_kernel<<<dim3(ntiles, B_SZ), 256, 0, stream>>>(
      H, logits, W_width, W_span, b_span, w_asp, b_asp, w_opn, b_opn,
      out + OFF_SR, out + OFF_ASP, out + OFF_OPN);
}